// EmotionalGraphNetwork_56023553409305
// MI455X (gfx1250) — compile-verified
//
#include <hip/hip_runtime.h>

// GCN on MI455X (gfx1250, wave32):
//   fp32 WMMA (v_wmma_f32_16x16x4_f32) for the dense GEMMs,
//   global fp32 atomics (L2-resident: agg buffers << 192MB L2) for segment sums.

typedef float v2f __attribute__((ext_vector_type(2)));
typedef float v8f __attribute__((ext_vector_type(8)));

#define IN_CH 256
#define HID1  128
#define HID2  64
#define N_CLS 8

// ---------------------------------------------------------------- utilities

__global__ void fill_kernel(float* __restrict__ p, float v, int n) {
    int i = blockIdx.x * blockDim.x + threadIdx.x;
    if (i < n) p[i] = v;
}

// deg[dst[e]] += ew[e]   (deg buffer pre-filled with 1.0 for the self-loop)
__global__ void deg_kernel(const int* __restrict__ dst, const float* __restrict__ ew,
                           float* __restrict__ deg, int E) {
    int e = blockIdx.x * blockDim.x + threadIdx.x;
    if (e < E) unsafeAtomicAdd(&deg[dst[e]], ew[e]);
}

// in-place deg -> deg^{-1/2}  (deg >= 1 always, so no zero guard needed)
__global__ void rsqrt_kernel(float* __restrict__ d, int n) {
    int i = blockIdx.x * blockDim.x + threadIdx.x;
    if (i < n) d[i] = rsqrtf(d[i]);
}

// norm[e] = dinv[src]*ew*dinv[dst]
__global__ void norm_kernel(const int* __restrict__ src, const int* __restrict__ dst,
                            const float* __restrict__ ew, const float* __restrict__ dinv,
                            float* __restrict__ norm, int E) {
    int e = blockIdx.x * blockDim.x + threadIdx.x;
    if (e < E) norm[e] = dinv[src[e]] * ew[e] * dinv[dst[e]];
}

// ------------------------------------------------------- WMMA fp32 GEMM C=A*B
// A: [M x K] row-major, B: [K x NN] row-major, C: [M x NN] row-major.
// One wave computes one 16x16 C tile; K-loop in steps of 4 via
// v_wmma_f32_16x16x4_f32.
//
// Fragment layouts (ISA 7.12.2, wave32):
//   A 16x4:  lanes 0-15 -> rows M=0..15, VGPR0/1 = K=0,1 ; lanes 16-31 -> K=2,3
//   B 4x16:  lanes 0-15 -> cols N=0..15, VGPR0/1 = K=0,1 ; lanes 16-31 -> K=2,3
//   C 16x16: lanes 0-15 -> N=0..15 rows M=r (VGPR r) ; lanes 16-31 -> rows M=8+r
template <int K, int NN>
__global__ __launch_bounds__(256) void gemm_wmma_f32(const float* __restrict__ A,
                                                     const float* __restrict__ B,
                                                     float* __restrict__ C, int M) {
    const int lane   = threadIdx.x & 31;
    const int wave   = threadIdx.x >> 5;
    const int ntiles = NN / 16;
    const int tilesM = (M + 15) >> 4;
    const int gw     = blockIdx.x * (blockDim.x >> 5) + wave;
    if (gw >= tilesM * ntiles) return;          // uniform per wave: EXEC stays all-1s

    const int tm   = gw / ntiles;
    const int tn   = gw - tm * ntiles;
    const int half = lane >> 4;                 // 0: K=0,1   1: K=2,3
    const int l15  = lane & 15;

    int arow = tm * 16 + l15;
    if (arow >= M) arow = M - 1;                // clamp reads; tail stores guarded
    const int bcol = tn * 16 + l15;

    const float* __restrict__ Arow = A + (size_t)arow * K;

    v8f acc = {};
#pragma unroll 4
    for (int k = 0; k < K; k += 4) {
        v2f a = *(const v2f*)(Arow + k + half * 2);
        v2f b;
        b.x = B[(size_t)(k + half * 2 + 0) * NN + bcol];
        b.y = B[(size_t)(k + half * 2 + 1) * NN + bcol];
        acc = __builtin_amdgcn_wmma_f32_16x16x4_f32(
            /*neg_a=*/false, a, /*neg_b=*/false, b,
            /*c_mod=*/(short)0, acc, /*reuse_a=*/false, /*reuse_b=*/false);
    }

    float* __restrict__ Cout = C + (size_t)(tm * 16 + half * 8) * NN + bcol;
    if (tm * 16 + 16 <= M) {
        // full tile (always taken for M % 16 == 0): 8 coalesced row stores
#pragma unroll
        for (int r = 0; r < 8; ++r) Cout[(size_t)r * NN] = acc[r];
    } else {
        // tail tile: guard each row
#pragma unroll
        for (int r = 0; r < 8; ++r) {
            if (tm * 16 + half * 8 + r < M) Cout[(size_t)r * NN] = acc[r];
        }
    }
}

// ------------------------------------------- edge gather + atomic scatter-add
// agg[dst[e], :] += norm[e] * h[src[e], :]
// C/4 lanes cooperate per edge; each handles a float4 -> coalesced 512B (C=128)
// or 256B (C=64) row transfers. agg fits in L2, so atomics are L2-resident.
template <int C>
__global__ void scatter_kernel(const int* __restrict__ src, const int* __restrict__ dst,
                               const float* __restrict__ norm, const float* __restrict__ h,
                               float* __restrict__ agg, int E) {
    const int  lpe   = C / 4;
    const long long total = (long long)E * lpe;
    const long long tid   = (long long)blockIdx.x * blockDim.x + threadIdx.x;
    if (tid >= total) return;
    const int e  = (int)(tid / lpe);
    const int c4 = (int)(tid % lpe) * 4;

    const float  w  = norm[e];
    const float4 hv = *(const float4*)(h + (size_t)src[e] * C + c4);
    float* outp = agg + (size_t)dst[e] * C + c4;
    unsafeAtomicAdd(outp + 0, w * hv.x);
    unsafeAtomicAdd(outp + 1, w * hv.y);
    unsafeAtomicAdd(outp + 2, w * hv.z);
    unsafeAtomicAdd(outp + 3, w * hv.w);
}

// agg[i,c] = (relu?)(agg[i,c] + dinv[i]^2 * h[i,c] + b[c])
template <int C, bool RELU>
__global__ void finish_kernel(float* __restrict__ agg, const float* __restrict__ h,
                              const float* __restrict__ dinv, const float* __restrict__ b,
                              int N) {
    const long long tid = (long long)blockIdx.x * blockDim.x + threadIdx.x;
    if (tid >= (long long)N * C) return;
    const int node = (int)(tid / C);
    const int c    = (int)(tid % C);
    const float di = dinv[node];
    float v = agg[tid] + di * di * h[tid] + b[c];
    agg[tid] = RELU ? fmaxf(v, 0.0f) : v;
}

// out[i,c] = sum_k h[i,k]*Wc[k,c] + bc[c]   (64x8 head: tiny, plain VALU)
__global__ void head_kernel(const float* __restrict__ h, const float* __restrict__ Wc,
                            const float* __restrict__ bc, float* __restrict__ out, int N) {
    const int i = blockIdx.x * blockDim.x + threadIdx.x;
    if (i >= N * N_CLS) return;
    const int node = i >> 3;
    const int c    = i & 7;
    const float* __restrict__ hr = h + (size_t)node * HID2;
    float s = bc[c];
#pragma unroll 8
    for (int k = 0; k < HID2; ++k) s += hr[k] * Wc[k * N_CLS + c];
    out[i] = s;
}

// ------------------------------------------------------------------- launcher

extern "C" void kernel_launch(void* const* d_in, const int* in_sizes, int n_in,
                              void* d_out, int out_size, void* d_ws, size_t ws_size,
                              hipStream_t stream) {
    const float* x    = (const float*)d_in[0];
    const int*   ei   = (const int*)d_in[1];   // [2,E] flat
    const float* ew   = (const float*)d_in[2];
    const float* W1   = (const float*)d_in[3];
    const float* b1   = (const float*)d_in[4];
    const float* W2   = (const float*)d_in[5];
    const float* b2   = (const float*)d_in[6];
    const float* Wc   = (const float*)d_in[7];
    const float* bc   = (const float*)d_in[8];
    float*       out  = (float*)d_out;

    const int N = in_sizes[0] / IN_CH;
    const int E = in_sizes[2];
    const int* src = ei;
    const int* dst = ei + E;

    // workspace carve-up (floats): dinv | norm | h1 | agg1 | h2 | agg2
    float* ws   = (float*)d_ws;
    float* dinv = ws;
    float* norm = dinv + (((size_t)N + 255) & ~(size_t)255);
    float* h1   = norm + (((size_t)E + 255) & ~(size_t)255);
    float* agg1 = h1   + (size_t)N * HID1;
    float* h2   = agg1 + (size_t)N * HID1;
    float* agg2 = h2   + (size_t)N * HID2;

    const int B = 256;
    auto blocks   = [&](long long n) { return (unsigned)((n + B - 1) / B); };
    auto gemmGrid = [&](int M, int NN) {
        long long waves = (long long)((M + 15) / 16) * (NN / 16);
        return (unsigned)((waves + 7) / 8);    // 8 waves / 256-thread block
    };

    // --- degree + normalization coefficients -------------------------------
    fill_kernel<<<blocks(N), B, 0, stream>>>(dinv, 1.0f, N);                 // self-loop
    deg_kernel<<<blocks(E), B, 0, stream>>>(dst, ew, dinv, E);
    rsqrt_kernel<<<blocks(N), B, 0, stream>>>(dinv, N);
    norm_kernel<<<blocks(E), B, 0, stream>>>(src, dst, ew, dinv, norm, E);

    // --- layer 1: h1 = x@W1 ; agg1 = scatter + self-loop ; relu ------------
    gemm_wmma_f32<IN_CH, HID1><<<gemmGrid(N, HID1), B, 0, stream>>>(x, W1, h1, N);
    fill_kernel<<<blocks((long long)N * HID1), B, 0, stream>>>(agg1, 0.0f, N * HID1);
    scatter_kernel<HID1><<<blocks((long long)E * (HID1 / 4)), B, 0, stream>>>(
        src, dst, norm, h1, agg1, E);
    finish_kernel<HID1, true><<<blocks((long long)N * HID1), B, 0, stream>>>(
        agg1, h1, dinv, b1, N);

    // --- layer 2: h2 = agg1@W2 ; agg2 = scatter + self-loop ----------------
    gemm_wmma_f32<HID1, HID2><<<gemmGrid(N, HID2), B, 0, stream>>>(agg1, W2, h2, N);
    fill_kernel<<<blocks((long long)N * HID2), B, 0, stream>>>(agg2, 0.0f, N * HID2);
    scatter_kernel<HID2><<<blocks((long long)E * (HID2 / 4)), B, 0, stream>>>(
        src, dst, norm, h2, agg2, E);
    finish_kernel<HID2, false><<<blocks((long long)N * HID2), B, 0, stream>>>(
        agg2, h2, dinv, b2, N);

    // --- classifier head ---------------------------------------------------
    head_kernel<<<blocks((long long)N * N_CLS), B, 0, stream>>>(agg2, Wc, bc, out, N);
}